// RoINetwork_37761352466883
// MI455X (gfx1250) — compile-verified
//
#include <hip/hip_runtime.h>
#include <math.h>

#define W_IMG 2048.0f
#define H_IMG 1024.0f
constexpr int BATCH = 8;
constexpr int G = 256;
constexpr int R = 16384;
constexpr int N = R + G;          // 16640 (divisible by 128)
constexpr int NUM_POS_MAX = 32;
constexpr int NUM_TOTAL = 128;

typedef unsigned int v4u __attribute__((ext_vector_type(4)));
typedef int          v4i __attribute__((ext_vector_type(4)));
typedef int          v8i __attribute__((ext_vector_type(8)));
typedef float        v2f __attribute__((ext_vector_type(2)));
typedef float        v8f __attribute__((ext_vector_type(8)));

__device__ __forceinline__ float4 load_allroi(const float4* rois4, const float4* gt4,
                                              int b, int row) {
  if (row < R) return rois4[(size_t)b * R + row];
  float4 g = gt4[(size_t)b * G + (row - R)];
  return make_float4(g.x / W_IMG, g.y / H_IMG, g.z / W_IMG, g.w / H_IMG);
}

// ---------------------------------------------------------------------------
// Phase 1: fused IoU row-max/argmax.  grid = (N/128, B), block = 256 (8 waves)
// Each wave owns 16 allroi rows; GT boxes live in LDS (TDM async tensor load).
// a1[i]+a2[j] outer-sum per 16x16 tile comes from V_WMMA_F32_16X16X4_F32.
// ---------------------------------------------------------------------------
__global__ void __launch_bounds__(256) phase1_iou(
    const float* __restrict__ gt, const float* __restrict__ rois,
    float* __restrict__ iomax, int* __restrict__ ioarg)
{
  __shared__ float4 s_gt[G];                 // 4 KB GT tile for this image
  const float4* gt4   = (const float4*)gt;
  const float4* rois4 = (const float4*)rois;
  const int b    = blockIdx.y;
  const int lane = threadIdx.x & 31;
  const int wave = threadIdx.x >> 5;
  const int hlf  = lane >> 4;                // which half of the wave
  const int col0 = lane & 15;
  const int rowBase = blockIdx.x * 128 + wave * 16;

  // ---- TDM: async tensor load of this image's GT boxes (4 KB) into LDS ----
  if (threadIdx.x == 0) {
    unsigned long long ga = (unsigned long long)(uintptr_t)(gt + (size_t)b * G * 4);
    unsigned ldsOff = (unsigned)(uintptr_t)(&s_gt[0]);    // low 32 bits = LDS byte offset
    v4u g0;
    g0.x = 1u;                                             // count = 1 descriptor
    g0.y = ldsOff;                                         // lds_addr
    g0.z = (unsigned)(ga & 0xFFFFFFFFull);                 // global_addr[31:0]
    g0.w = (unsigned)((ga >> 32) & 0x1FFFFFFull) | (2u << 30); // addr[56:32] | type=2
    v8i g1;
    g1[0] = 0x00020000;                                    // data_size = 4 B
    g1[1] = (int)(((unsigned)(G * 4) & 0xFFFFu) << 16);    // tensor_dim0 = 1024 elems
    g1[2] = (int)(1u << 16);                               // tensor_dim1 = 1
    g1[3] = (int)(((unsigned)(G * 4) & 0xFFFFu) << 16);    // tile_dim0 = 1024
    g1[4] = 1;                                             // tile_dim1 = 1, tile_dim2 = 0
    g1[5] = G * 4;                                         // tensor_dim0_stride
    g1[6] = (int)(((unsigned)(G * 4) & 0xFFFFu) << 16);    // tensor_dim1_stride lo16
    g1[7] = 0;
    v4i z4 = {0, 0, 0, 0};
#if __has_builtin(__builtin_amdgcn_tensor_load_to_lds)
# if defined(__clang_major__) && __clang_major__ >= 23
    v8i z8 = {0, 0, 0, 0, 0, 0, 0, 0};
    __builtin_amdgcn_tensor_load_to_lds(g0, g1, z4, z4, z8, 0);
# else
    __builtin_amdgcn_tensor_load_to_lds(g0, g1, z4, z4, 0);
# endif
#endif
#if __has_builtin(__builtin_amdgcn_s_wait_tensorcnt)
    __builtin_amdgcn_s_wait_tensorcnt(0);
#endif
  }
  // Safety net: same bytes via vector loads (keeps LDS contents well-defined
  // for the compiler; identical data, benign overlap with the TDM write).
  for (int i = threadIdx.x; i < G; i += blockDim.x)
    s_gt[i] = gt4[(size_t)b * G + i];
  __syncthreads();

  // Per-lane register tile: 8 rows (this half's rows rowBase + 8*hlf + v)
  float rx1[8], ry1[8], rx2[8], ry2[8], ra[8];
#pragma unroll
  for (int v = 0; v < 8; ++v) {
    float4 bb = load_allroi(rois4, gt4, b, rowBase + hlf * 8 + v);
    rx1[v] = bb.x; ry1[v] = bb.y; rx2[v] = bb.z; ry2[v] = bb.w;
    ra[v]  = fmaxf((bb.z - bb.x) * (bb.w - bb.y), 0.0f);
  }
  // WMMA A matrix (16x4 f32): lanes 0-15 hold row M=lane -> {a1[M], 1};
  // lanes 16-31 hold K=2,3 of row (lane-16) -> zeros (K padding).
  float4 abb = load_allroi(rois4, gt4, b, rowBase + col0);
  float a1A = fmaxf((abb.z - abb.x) * (abb.w - abb.y), 0.0f);
  v2f Amat;
  Amat.x = hlf ? 0.0f : a1A;
  Amat.y = hlf ? 0.0f : 1.0f;

  float best[8]; int barg[8];
#pragma unroll
  for (int v = 0; v < 8; ++v) { best[v] = -__builtin_inff(); barg[v] = 0; }

  for (int t = 0; t < 16; ++t) {             // 16 GT tiles of 16 columns
    const int col = t * 16 + col0;
    float4 gb = s_gt[col];
    float a2 = fmaxf((gb.z - gb.x) * (gb.w - gb.y), 0.0f);
    v8f suma;
#if __has_builtin(__builtin_amdgcn_wmma_f32_16x16x4_f32)
    // B matrix (4x16 f32): row K=0 = ones, row K=1 = a2[col]; K=2,3 padding.
    v2f Bmat;
    Bmat.x = hlf ? 0.0f : 1.0f;
    Bmat.y = hlf ? 0.0f : a2;
    v8f cz = {0.f, 0.f, 0.f, 0.f, 0.f, 0.f, 0.f, 0.f};
    suma = __builtin_amdgcn_wmma_f32_16x16x4_f32(
        false, Amat, false, Bmat, (short)0, cz, false, false);
#else
#pragma unroll
    for (int v = 0; v < 8; ++v) suma[v] = ra[v] + a2;   // VALU fallback
#endif
#pragma unroll
    for (int v = 0; v < 8; ++v) {            // D layout: M = v + 8*hlf, N = col
      float lbx = fmaxf(rx1[v], gb.x), lby = fmaxf(ry1[v], gb.y);
      float ubx = fminf(rx2[v], gb.z), uby = fminf(ry2[v], gb.w);
      float wi = fmaxf(ubx - lbx, 0.0f), ih = fmaxf(uby - lby, 0.0f);
      float inter = wi * ih;
      float iou = inter / (suma[v] - inter);
      if (iou > best[v]) { best[v] = iou; barg[v] = col; }
    }
  }

  // Reduce max/argmax across the 16 columns (lanes within each half-wave).
#pragma unroll
  for (int v = 0; v < 8; ++v) {
#pragma unroll
    for (int m = 1; m < 16; m <<= 1) {
      float ov = __shfl_xor(best[v], m, 32);
      int   oa = __shfl_xor(barg[v], m, 32);
      if (ov > best[v] || (ov == best[v] && oa < barg[v])) { best[v] = ov; barg[v] = oa; }
    }
  }
  if (col0 == 0) {
#pragma unroll
    for (int v = 0; v < 8; ++v) {
      int row = rowBase + hlf * 8 + v;
      iomax[(size_t)b * N + row] = best[v];
      ioarg[(size_t)b * N + row] = barg[v];
    }
  }
}

// ---------------------------------------------------------------------------
// Phase 2: exact rank sampling + gather + encode.  grid = B, block = 1024.
// ---------------------------------------------------------------------------
__global__ void __launch_bounds__(1024) phase2_sample(
    const float* __restrict__ gt, const int* __restrict__ gtlab,
    const float* __restrict__ rois,
    const float* __restrict__ rpos, const float* __restrict__ rneg,
    const float* __restrict__ iomax, const int* __restrict__ ioarg,
    float* __restrict__ out)
{
  const int b = blockIdx.x;
  const int t = threadIdx.x;
  __shared__ int s_c;
  __shared__ unsigned long long s_sel[NUM_TOTAL];
  __shared__ int s_ids[NUM_TOTAL];

  const float* im = iomax + (size_t)b * N;
  const float* rp = rpos + (size_t)b * N;
  const float* rn = rneg + (size_t)b * N;

  if (t < NUM_TOTAL) s_ids[t] = 0;           // drop-mode scatter default
  if (t == 0) s_c = 0;
  __syncthreads();
  int lc = 0;
  for (int i = t; i < N; i += 1024) lc += (im[i] >= 0.5f) ? 1 : 0;
  if (lc) atomicAdd(&s_c, lc);
  __syncthreads();
  const int numpos = min(s_c, NUM_POS_MAX);
  const int numneg = NUM_TOTAL - numpos;

  for (int pass = 0; pass < 2; ++pass) {
    const int K = pass ? numneg : numpos;
    const int base = pass ? numpos : 0;
    const float* keys = pass ? rn : rp;
    auto packed = [&](int i) -> unsigned long long {
      float v = im[i];
      bool m = pass ? ((v < 0.5f) && (v >= 0.0f)) : (v >= 0.5f);
      if (!m) return ~0ull;
      return ((unsigned long long)__float_as_uint(keys[i]) << 32) | (unsigned)i;
    };
    // K-th smallest packed key via bitwise binary search (block-wide counts).
    unsigned long long lo = 0, hi = ~0ull;
    while (lo < hi) {
      unsigned long long mid = lo + ((hi - lo) >> 1);
      __syncthreads();
      if (t == 0) s_c = 0;
      __syncthreads();
      int c = 0;
      for (int i = t; i < N; i += 1024) c += (packed(i) <= mid) ? 1 : 0;
      if (c) atomicAdd(&s_c, c);
      __syncthreads();
      if (s_c >= K) hi = mid; else lo = mid + 1;
    }
    const unsigned long long V = lo;
    __syncthreads();
    if (t == 0) s_c = 0;
    __syncthreads();
    for (int i = t; i < N; i += 1024) {
      unsigned long long p = packed(i);
      if (p != ~0ull && p <= V) {
        int slot = atomicAdd(&s_c, 1);
        if (slot < NUM_TOTAL) s_sel[slot] = p;
      }
    }
    __syncthreads();
    int kact = min(s_c, NUM_TOTAL);
    if (t < kact) {                          // exact rank among survivors
      unsigned long long mine = s_sel[t];
      int r = 0;
      for (int j = 0; j < kact; ++j) r += (s_sel[j] < mine) ? 1 : 0;
      int slot = base + r;
      if (slot < NUM_TOTAL) s_ids[slot] = (int)(mine & 0xFFFFFFFFull);
    }
    __syncthreads();
  }

  if (t < NUM_TOTAL) {
    const float4* gt4   = (const float4*)gt;
    const float4* rois4 = (const float4*)rois;
    int id = s_ids[t];
    int am = ioarg[(size_t)b * N + id];
    float lab = (t < numpos) ? (float)(gtlab[b * G + am] + 1) : 0.0f;
    float4 tb = gt4[(size_t)b * G + am];
    float4 sr;
    if (id < R) sr = rois4[(size_t)b * R + id];
    else {
      float4 g = gt4[(size_t)b * G + (id - R)];
      sr = make_float4(g.x / W_IMG, g.y / H_IMG, g.z / W_IMG, g.w / H_IMG);
    }
    // faithful halved-wh cxcywh + encode / std
    float gcx = (tb.x + tb.z) * 0.5f, gcy = (tb.y + tb.w) * 0.5f;
    float gw  = (tb.z - tb.x) * 0.5f, gh  = (tb.w - tb.y) * 0.5f;
    float acx = (sr.x + sr.z) * 0.5f, acy = (sr.y + sr.w) * 0.5f;
    float aw  = (sr.z - sr.x) * 0.5f, ah  = (sr.w - sr.y) * 0.5f;
    float e0 = ((gcx - acx) / aw) / 0.1f;
    float e1 = ((gcy - acy) / ah) / 0.1f;
    float e2 = logf(gw / aw) / 0.2f;
    float e3 = logf(gh / ah) / 0.2f;
    int k = b * NUM_TOTAL + t;
    out[k] = lab;                                        // labels (B,128)
    float* enc = out + BATCH * NUM_TOTAL;                // encoded (B,128,4)
    enc[k * 4 + 0] = e0; enc[k * 4 + 1] = e1; enc[k * 4 + 2] = e2; enc[k * 4 + 3] = e3;
    float* sro = out + BATCH * NUM_TOTAL + BATCH * NUM_TOTAL * 4;  // rois (B,128,4)
    sro[k * 4 + 0] = sr.x; sro[k * 4 + 1] = sr.y; sro[k * 4 + 2] = sr.z; sro[k * 4 + 3] = sr.w;
  }
}

extern "C" void kernel_launch(void* const* d_in, const int* in_sizes, int n_in,
                              void* d_out, int out_size, void* d_ws, size_t ws_size,
                              hipStream_t stream) {
  (void)in_sizes; (void)n_in; (void)out_size; (void)ws_size;
  const float* gt   = (const float*)d_in[0];
  const int*   gl   = (const int*)d_in[1];
  const float* rois = (const float*)d_in[2];
  const float* rp   = (const float*)d_in[3];
  const float* rn   = (const float*)d_in[4];
  float* out = (float*)d_out;
  float* iomax = (float*)d_ws;                                        // B*N floats
  int*   ioarg = (int*)((char*)d_ws + (size_t)BATCH * N * sizeof(float)); // B*N ints

  dim3 grid1(N / 128, BATCH);
  phase1_iou<<<grid1, 256, 0, stream>>>(gt, rois, iomax, ioarg);
  phase2_sample<<<BATCH, 1024, 0, stream>>>(gt, gl, rois, rp, rn, iomax, ioarg, out);
}